// intra_att_58102317580556
// MI455X (gfx1250) — compile-verified
//
#include <hip/hip_runtime.h>
#include <hip/hip_bf16.h>
#include <math.h>

#define KN 32
#define DIM 128
#define EMB_STRIDE 132   // floats: 528B row stride -> 16B aligned, <=2-way LDS bank conflicts

#if defined(__gfx1250__) && __has_builtin(__builtin_amdgcn_global_load_async_to_lds_b128)
#define HAVE_ASYNC_LDS 1
#else
#define HAVE_ASYNC_LDS 0
#endif

// Vector + address-space typed pointers matching the builtin's signature:
//   __builtin_amdgcn_global_load_async_to_lds_b128(v4i AS1* src, v4i AS3* dst, imm, imm)
typedef int v4i __attribute__((vector_size(16)));
typedef __attribute__((address_space(1))) v4i  g_v4i;
typedef __attribute__((address_space(3))) v4i  l_v4i;

// Copy 16 bytes global -> LDS. CDNA5 path: async DMA tracked by ASYNCcnt.
__device__ __forceinline__ void async_copy16(const float* gsrc, float* ldst) {
#if HAVE_ASYNC_LDS
    __builtin_amdgcn_global_load_async_to_lds_b128(
        (g_v4i*)gsrc,      // addrspacecast generic -> global, drops const
        (l_v4i*)ldst,      // addrspacecast generic -> LDS
        0, 0);
#else
    *(float4*)ldst = *(const float4*)gsrc;
#endif
}

__device__ __forceinline__ void wait_async_lds() {
#if HAVE_ASYNC_LDS && __has_builtin(__builtin_amdgcn_s_wait_asynccnt)
    __builtin_amdgcn_s_wait_asynccnt(0);
#endif
}

__global__ __launch_bounds__(128)
void intra_att_kernel(const int* __restrict__ nei,
                      const float* __restrict__ h,
                      const float* __restrict__ h_refer,
                      const float* __restrict__ att,
                      float* __restrict__ out)
{
    __shared__ __align__(16) float  emb[KN * EMB_STRIDE];  // 16.9 KB gathered tile
    __shared__ __align__(16) float  att_nei_s[DIM];
    __shared__ __align__(16) float  part[4 * KN];          // score partials [wave][k]
    __shared__ __align__(16) float4 opart[4 * KN];         // output partials [wave][lane]
    __shared__ __align__(16) float  wgt[KN];
    __shared__ float base_part[4];

    const int n    = blockIdx.x;
    const int t    = threadIdx.x;
    const int w    = t >> 5;             // wave id 0..3
    const int lane = t & 31;

    // ---- Stage 1 (issued FIRST so the async DMA overlaps stage 0):
    // wave w owns rows {w, w+4, ..., w+28}. Lane groups of 8 fetch the 8 row
    // indices, broadcast with shfl, then each lane async-copies 16B of the row.
    const int myidx = nei[n * KN + w + ((lane & 7) << 2)];
    #pragma unroll
    for (int i = 0; i < 8; ++i) {
        const int ridx = __shfl(myidx, i, 8);              // row (4i + w)'s index
        const int r    = (i << 2) + w;
        async_copy16(h + (size_t)ridx * DIM + (lane << 2),
                     &emb[r * EMB_STRIDE + (lane << 2)]);
    }

    // ---- Stage 0 (overlaps DMA): stage att_nei; base = dot(h_refer[n], att_ref)
    att_nei_s[t] = att[DIM + t];
    float bp = h_refer[(size_t)n * DIM + t] * att[t];
    #pragma unroll
    for (int off = 16; off > 0; off >>= 1) bp += __shfl_xor(bp, off, 32);
    if (lane == 0) base_part[w] = bp;

    wait_async_lds();
    __syncthreads();

    // ---- Stage 2: partial scores. lane = k, wave w covers dims [32w, 32w+32)
    {
        const int k = lane;
        const float4* row4 = (const float4*)&emb[k * EMB_STRIDE + (w << 5)];
        const float4* an4  = (const float4*)&att_nei_s[w << 5];
        float acc = 0.f;
        #pragma unroll
        for (int j = 0; j < 8; ++j) {
            float4 r = row4[j], a = an4[j];
            acc = fmaf(r.x, a.x, acc);
            acc = fmaf(r.y, a.y, acc);
            acc = fmaf(r.z, a.z, acc);
            acc = fmaf(r.w, a.w, acc);
        }
        part[(w << 5) + k] = acc;
    }
    __syncthreads();

    // ---- Stage 3: wave 0 finishes scores + leaky-relu + softmax (wave32 shuffles)
    if (w == 0) {
        const int k = lane;
        const float base = base_part[0] + base_part[1] + base_part[2] + base_part[3];
        float s = base + part[k] + part[32 + k] + part[64 + k] + part[96 + k];
        s = (s > 0.f) ? s : 0.01f * s;                     // leaky_relu(0.01)
        float m = s;
        #pragma unroll
        for (int off = 16; off > 0; off >>= 1) m = fmaxf(m, __shfl_xor(m, off, 32));
        const float e = __expf(s - m);
        float sum = e;
        #pragma unroll
        for (int off = 16; off > 0; off >>= 1) sum += __shfl_xor(sum, off, 32);
        wgt[k] = e / sum;
    }
    __syncthreads();

    // ---- Stage 4: out[n,:] = sum_k w[k] * emb[k,:]
    // Wave w covers k in [8w, 8w+8); lane accumulates dims [4*lane, 4*lane+4)
    // via conflict-free ds_load_b128 row reads; weights via 2 broadcast b128 loads.
    float4 acc = make_float4(0.f, 0.f, 0.f, 0.f);
    {
        const float4* wg4 = (const float4*)&wgt[w << 3];
        const float4 wa = wg4[0], wb = wg4[1];
        const float wr[8] = {wa.x, wa.y, wa.z, wa.w, wb.x, wb.y, wb.z, wb.w};
        #pragma unroll
        for (int j = 0; j < 8; ++j) {
            const int k = (w << 3) + j;
            const float4 e = *(const float4*)&emb[k * EMB_STRIDE + (lane << 2)];
            acc.x = fmaf(wr[j], e.x, acc.x);
            acc.y = fmaf(wr[j], e.y, acc.y);
            acc.z = fmaf(wr[j], e.z, acc.z);
            acc.w = fmaf(wr[j], e.w, acc.w);
        }
    }
    opart[(w << 5) + lane] = acc;
    __syncthreads();

    if (w == 0) {
        const float4 a = opart[lane];
        const float4 b = opart[32 + lane];
        const float4 c = opart[64 + lane];
        const float4 d = opart[96 + lane];
        float4 r;
        r.x = (a.x + b.x) + (c.x + d.x);
        r.y = (a.y + b.y) + (c.y + d.y);
        r.z = (a.z + b.z) + (c.z + d.z);
        r.w = (a.w + b.w) + (c.w + d.w);
        *(float4*)&out[(size_t)n * DIM + (lane << 2)] = r;  // coalesced b128 store
    }
}

extern "C" void kernel_launch(void* const* d_in, const int* in_sizes, int n_in,
                              void* d_out, int out_size, void* d_ws, size_t ws_size,
                              hipStream_t stream) {
    const int*   nei = (const int*)d_in[0];   // (N, 32) indices
    const float* h   = (const float*)d_in[1]; // (M, 128)
    const float* hr  = (const float*)d_in[2]; // (N, 128)
    const float* att = (const float*)d_in[3]; // (1, 256)
    float* out = (float*)d_out;               // (N, 128)

    const int N = in_sizes[0] / KN;
    intra_att_kernel<<<N, 128, 0, stream>>>(nei, h, hr, att, out);
}